// CombSubFastFacV1_66821101191257
// MI455X (gfx1250) — compile-verified
//
#include <hip/hip_runtime.h>
#include <math.h>

// ---------------------------------------------------------------------------
// CombSub synth for MI455X (gfx1250, wave32).
//   - 1024-pt FFT as four-step 32x32: two complex 32x32x32 matmuls on the
//     matrix pipe via V_WMMA_F32_16X16X4_F32 (fp32-exact).
//   - One wave per frame; complex data staged in LDS (stride 33, no bank
//     conflicts); DFT/twiddle factors computed analytically per lane directly
//     in the WMMA A/B fragment layouts (no constant memory traffic).
//   - Combtooth phase = 64KB exclusive block-prefix table + closed form, so
//     the whole pipeline is fused: only filters/noise in, signal out.
// ---------------------------------------------------------------------------

namespace {
constexpr int   kBatch  = 8;
constexpr int   kBlocks = 2048;           // f0 frames / output blocks
constexpr int   kBS     = 512;
constexpr int   kNS     = kBlocks * kBS;  // 1048576 samples per batch
constexpr int   kFrames = kBlocks + 1;    // 2049 analysis frames
constexpr int   kN      = 1024;           // FFT length
constexpr int   kBins   = 513;
constexpr float kSR     = 44100.0f;
}

typedef float v2f __attribute__((ext_vector_type(2)));
typedef float v8f __attribute__((ext_vector_type(8)));

__device__ __forceinline__ v8f wmma4(v2f a, v2f b, v8f c) {
  // D = A(16x4) * B(4x16) + C, fp32. 8-arg form: (neg_a,A,neg_b,B,c_mod,C,reuse_a,reuse_b)
  return __builtin_amdgcn_wmma_f32_16x16x4_f32(false, a, false, b, (short)0, c,
                                               false, false);
}

// ---- W32[m][k] = cos/sin(2*pi*m*k/32), emitted directly in fragment layout.
// A fragment (16x4 tile, ISA 7.12.2): lanes 0-15 row m=L, {K,K+1}; lanes 16-31
// row m=L-16, {K+2,K+3}.  qc = global K chunk (0..7) of the 32-deep dimension.
__device__ __forceinline__ v2f wfragA(int r, int qc, int part) {
  int L  = threadIdx.x;
  int m  = 16 * r + (L & 15);
  int kb = 4 * qc + ((L >> 4) << 1);
  float a0 = (float)((m * kb) & 31) * (1.0f / 16.0f);
  float a1 = (float)((m * (kb + 1)) & 31) * (1.0f / 16.0f);
  v2f o;
  o.x = part ? sinpif(a0) : cospif(a0);
  o.y = part ? sinpif(a1) : cospif(a1);
  return o;
}

__device__ __forceinline__ v2f wfragB(int qc, int c, int part) {
  int L  = threadIdx.x;
  int n  = 16 * c + (L & 15);
  int kb = 4 * qc + ((L >> 4) << 1);
  float a0 = (float)((kb * n) & 31) * (1.0f / 16.0f);
  float a1 = (float)(((kb + 1) * n) & 31) * (1.0f / 16.0f);
  v2f o;
  o.x = part ? sinpif(a0) : cospif(a0);
  o.y = part ? sinpif(a1) : cospif(a1);
  return o;
}

// LDS (32x32 matrix, row stride 33) -> A / B fragments.
__device__ __forceinline__ v2f ldsA(const float* base, int r, int qc) {
  int L  = threadIdx.x;
  int m  = 16 * r + (L & 15);
  int kb = 4 * qc + ((L >> 4) << 1);
  v2f o;
  o.x = base[m * 33 + kb];
  o.y = base[m * 33 + kb + 1];
  return o;
}

__device__ __forceinline__ v2f ldsB(const float* base, int qc, int c) {
  int L  = threadIdx.x;
  int n  = 16 * c + (L & 15);
  int kb = 4 * qc + ((L >> 4) << 1);
  v2f o;
  o.x = base[kb * 33 + n];
  o.y = base[(kb + 1) * 33 + n];
  return o;
}

// Stage: Y = W32 * M (complex, sign SGN: -1 fwd, +1 inv), then column twiddle
// exp(SGN*2*pi*i*k2*n1/1024); done in-place on the LDS matrix (staged in regs).
template <int SGN>
__device__ void stageA_twiddle(float* ldre, float* ldim) {
  const int L = threadIdx.x;
  float sre[2][2][8], sim[2][2][8];
#pragma unroll
  for (int r = 0; r < 2; ++r) {
#pragma unroll
    for (int c = 0; c < 2; ++c) {
      v8f Pcc = {0, 0, 0, 0, 0, 0, 0, 0};
      v8f Pss = Pcc, Pcs = Pcc, Psc = Pcc;
#pragma unroll
      for (int qc = 0; qc < 8; ++qc) {
        v2f ac = wfragA(r, qc, 0);
        v2f as = wfragA(r, qc, 1);
        v2f br = ldsB(ldre, qc, c);
        v2f bi = ldsB(ldim, qc, c);
        Pcc = wmma4(ac, br, Pcc);
        Pss = wmma4(as, bi, Pss);
        Pcs = wmma4(ac, bi, Pcs);
        Psc = wmma4(as, br, Psc);
      }
#pragma unroll
      for (int j = 0; j < 8; ++j) {
        // (cos + i*SGN*sin)(Xr + i*Xi)
        float re = Pcc[j] - (float)SGN * Pss[j];
        float im = Pcs[j] + (float)SGN * Psc[j];
        int k2 = 16 * r + j + 8 * (L >> 4);
        int n1 = 16 * c + (L & 15);
        float t  = (float)((k2 * n1) & (kN - 1)) * (1.0f / 512.0f); // units of pi
        float cw = cospif(t);
        float sw = (float)SGN * sinpif(t);
        sre[r][c][j] = re * cw - im * sw;
        sim[r][c][j] = re * sw + im * cw;
      }
    }
  }
  __syncthreads();
#pragma unroll
  for (int r = 0; r < 2; ++r)
#pragma unroll
    for (int c = 0; c < 2; ++c)
#pragma unroll
      for (int j = 0; j < 8; ++j) {
        int row = 16 * r + j + 8 * (L >> 4);
        int col = 16 * c + (L & 15);
        ldre[row * 33 + col] = sre[r][c][j];
        ldim[row * 33 + col] = sim[r][c][j];
      }
  __syncthreads();
}

// ---------------------------------------------------------------------------
// K1: per-batch exclusive prefix of per-block phase sums.
// Block i sum = (256.5*f0[i] + 255.5*f0[i+1]) / sr   (piecewise-linear f0).
// ---------------------------------------------------------------------------
__global__ __launch_bounds__(256) void scan_kernel(const float* __restrict__ f0,
                                                   float* __restrict__ P) {
  __shared__ float sh[256];
  const int b = blockIdx.x;
  const int t = threadIdx.x;
  const float inv_sr = 1.0f / kSR;
  float carry = 0.0f;
  for (int ch = 0; ch < kBlocks / 256; ++ch) {
    int i = ch * 256 + t;
    float fa = f0[b * kBlocks + i];
    float fb = f0[b * kBlocks + min(i + 1, kBlocks - 1)];
    float s  = (256.5f * fa + 255.5f * fb) * inv_sr;
    sh[t] = s;
    __syncthreads();
    float v = s;
    for (int off = 1; off < 256; off <<= 1) {
      float u = (t >= off) ? sh[t - off] : 0.0f;
      __syncthreads();
      v += u;
      sh[t] = v;
      __syncthreads();
    }
    P[b * kBlocks + i] = carry + v - s;  // exclusive prefix
    carry += sh[255];
    __syncthreads();
  }
}

__global__ __launch_bounds__(256) void zero_kernel(float* __restrict__ out, int n) {
  int i = blockIdx.x * blockDim.x + threadIdx.x;
  int stride = gridDim.x * blockDim.x;
  for (; i < n; i += stride) out[i] = 0.0f;
}

// ---------------------------------------------------------------------------
// K3: fused frame pipeline.  One wave32 per (batch, frame).
// ---------------------------------------------------------------------------
__global__ __launch_bounds__(32) void synth_kernel(
    const float* __restrict__ f0, const float* __restrict__ hm,
    const float* __restrict__ hp, const float* __restrict__ nm,
    const float* __restrict__ nu, const float* __restrict__ P,
    float* __restrict__ out) {
  __shared__ float ldre[32 * 33];
  __shared__ float ldim[32 * 33];
  __shared__ float zre[kN];
  __shared__ float zim[kN];

  const int L = threadIdx.x;
  const int i = blockIdx.x;  // frame 0..2048
  const int b = blockIdx.y;

  const float* f0b = f0 + b * kBlocks;
  const float* Pb  = P + b * kBlocks;
  const float* nub = nu + (size_t)b * kNS;

  // ---- generate windowed comb (re) + noise (im), packed complex ----
#pragma unroll 4
  for (int it = 0; it < 32; ++it) {
    int n = it * 32 + L;
    int g = (i - 1) * kBS + n;  // index into the un-padded signal
    float cv = 0.0f, nv = 0.0f;
    if (g >= 0 && g < kNS) {
      int ib = g >> 9;
      int jj = g & 511;
      float fa  = f0b[ib];
      float fb2 = f0b[min(ib + 1, kBlocks - 1)];
      float fup = fa + (fb2 - fa) * ((float)jj * (1.0f / 512.0f));
      float part = ((float)(jj + 1) * fa +
                    (float)(jj * (jj + 1) / 2) * (fb2 - fa) * (1.0f / 512.0f)) *
                   (1.0f / kSR);
      float x = Pb[ib] + part;
      x -= rintf(x);  // wrap (round-half-even like jnp.round)
      float targ = kSR * x / (fup + 0.001f);
      cv = (targ == 0.0f) ? 1.0f
                          : sinpif(targ) / (3.14159265358979f * targ);
      nv = nub[g] * 2.0f - 1.0f;
    }
    float w = sinpif((float)n * (1.0f / 1024.0f));  // sqrt-hann
    ldre[it * 33 + L] = cv * w;
    ldim[it * 33 + L] = nv * w;
  }
  __syncthreads();

  // ---- forward FFT: W*M + twiddle, then (.)*W -> Z[k2 + 32*k1] ----
  stageA_twiddle<-1>(ldre, ldim);
  {
#pragma unroll
    for (int r = 0; r < 2; ++r) {
#pragma unroll
      for (int c = 0; c < 2; ++c) {
        v8f Pcc = {0, 0, 0, 0, 0, 0, 0, 0};
        v8f Pss = Pcc, Pcs = Pcc, Psc = Pcc;
#pragma unroll
        for (int qc = 0; qc < 8; ++qc) {
          v2f ar = ldsA(ldre, r, qc);
          v2f ai = ldsA(ldim, r, qc);
          v2f bc = wfragB(qc, c, 0);
          v2f bs = wfragB(qc, c, 1);
          Pcc = wmma4(ar, bc, Pcc);
          Pss = wmma4(ai, bs, Pss);
          Pcs = wmma4(ai, bc, Pcs);
          Psc = wmma4(ar, bs, Psc);
        }
#pragma unroll
        for (int j = 0; j < 8; ++j) {
          float re = Pcc[j] + Pss[j];  // SGN=-1: Re = Pcc - SGN*Pss
          float im = Pcs[j] - Psc[j];  //         Im = Pcs + SGN*Psc
          int k2 = 16 * r + j + 8 * (L >> 4);
          int k1 = 16 * c + (L & 15);
          zre[k2 + 32 * k1] = re;
          zim[k2 + 32 * k1] = im;
        }
      }
    }
  }
  __syncthreads();

  // ---- Hermitian split, filter, build full conjugate-symmetric spectrum ----
  {
    int fi = min(i, kBlocks - 1);
    const float* hmb = hm + ((size_t)b * kBlocks + fi) * kBins;
    const float* hpb = hp + ((size_t)b * kBlocks + fi) * kBins;
    const float* nmb = nm + ((size_t)b * kBlocks + fi) * kBins;
#pragma unroll 1
    for (int kk = 0; kk < 17; ++kk) {
      int k = kk * 32 + L;
      if (k <= 512) {
        int km = (kN - k) & (kN - 1);
        float zr = zre[k], zi = zim[k];
        float mr = zre[km], mi = zim[km];
        float Cr = 0.5f * (zr + mr), Ci = 0.5f * (zi - mi);   // comb spectrum
        float Nr = 0.5f * (zi + mi), Ni = -0.5f * (zr - mr);  // noise spectrum
        float em = expf(hmb[k]);
        float ph = hpb[k];
        float Hr = em * cospif(ph), Hi = em * sinpif(ph);
        float G  = expf(nmb[k]) * (1.0f / 128.0f);
        float Sr = Cr * Hr - Ci * Hi + Nr * G;
        float Si = Cr * Hi + Ci * Hr + Ni * G;
        ldre[(k >> 5) * 33 + (k & 31)] = Sr;
        ldim[(k >> 5) * 33 + (k & 31)] = Si;
        if (k >= 1 && k <= 511) {
          int k2 = kN - k;
          ldre[(k2 >> 5) * 33 + (k2 & 31)] = Sr;
          ldim[(k2 >> 5) * 33 + (k2 & 31)] = -Si;
        }
      }
    }
  }
  __syncthreads();

  // ---- inverse FFT (real output only in last stage), window, OLA ----
  stageA_twiddle<1>(ldre, ldim);
  {
    float* ob = out + (size_t)b * kNS;
#pragma unroll
    for (int r = 0; r < 2; ++r) {
#pragma unroll
      for (int c = 0; c < 2; ++c) {
        v8f Pcc = {0, 0, 0, 0, 0, 0, 0, 0};
        v8f Pss = Pcc;
#pragma unroll
        for (int qc = 0; qc < 8; ++qc) {
          v2f ar = ldsA(ldre, r, qc);
          v2f ai = ldsA(ldim, r, qc);
          v2f bc = wfragB(qc, c, 0);
          v2f bs = wfragB(qc, c, 1);
          Pcc = wmma4(ar, bc, Pcc);
          Pss = wmma4(ai, bs, Pss);
        }
#pragma unroll
        for (int j = 0; j < 8; ++j) {
          float re = (Pcc[j] - Pss[j]) * (1.0f / 1024.0f);  // SGN=+1 real part
          int k2 = 16 * r + j + 8 * (L >> 4);
          int k1 = 16 * c + (L & 15);
          int t = k2 + 32 * k1;
          float v = re * sinpif((float)t * (1.0f / 1024.0f));
          if (t < kBS) {
            if (i >= 1) atomicAdd(&ob[(size_t)(i - 1) * kBS + t], v);
          } else {
            if (i <= kBlocks - 1) atomicAdd(&ob[(size_t)i * kBS + (t - kBS)], v);
          }
        }
      }
    }
  }
}

extern "C" void kernel_launch(void* const* d_in, const int* in_sizes, int n_in,
                              void* d_out, int out_size, void* d_ws,
                              size_t ws_size, hipStream_t stream) {
  const float* f0 = (const float*)d_in[0];
  const float* hm = (const float*)d_in[1];
  const float* hp = (const float*)d_in[2];
  const float* nm = (const float*)d_in[3];
  const float* nu = (const float*)d_in[4];
  float* out = (float*)d_out;
  float* P   = (float*)d_ws;  // 8*2048 floats = 64 KB block-phase prefix

  hipLaunchKernelGGL(scan_kernel, dim3(kBatch), dim3(256), 0, stream, f0, P);
  hipLaunchKernelGGL(zero_kernel, dim3(4096), dim3(256), 0, stream, out,
                     kBatch * kNS);
  hipLaunchKernelGGL(synth_kernel, dim3(kFrames, kBatch), dim3(32), 0, stream,
                     f0, hm, hp, nm, nu, P, out);
}